// AlphaCompositor_49349174231582
// MI455X (gfx1250) — compile-verified
//
#include <hip/hip_runtime.h>
#include <stdint.h>

#define H_  512
#define W_  512
#define HW_ (H_ * W_)
#define N_  8
#define K_  16

typedef __attribute__((ext_vector_type(4))) unsigned int u32x4;
typedef __attribute__((ext_vector_type(8))) unsigned int u32x8;

// ---------------------------------------------------------------------------
// TDM: async 2D strided-tile load  Global -> LDS  (CDNA5 Tensor Data Mover)
// D# layout per CDNA5 ISA.
//   tile:   tile_d0 elements (contiguous, 4B each) x tile_d1 rows
//   memory: rows separated by stride_elems * 4 bytes
//   LDS:    rows packed back-to-back starting at lds_byte
// Loads are marked non-temporal: the frag/alpha stream (~256MB/pass) must not
// evict the L2-resident 3.2MB point table that the gathers depend on.
// ---------------------------------------------------------------------------
__device__ __forceinline__ void tdm_load_tile_2d(uint32_t lds_byte,
                                                 const void* gptr,
                                                 uint32_t tile_d0,
                                                 uint32_t tile_d1,
                                                 uint64_t stride_elems) {
  const uint64_t ga = (uint64_t)(uintptr_t)gptr;
  // Group 0: count=1, is_restore=0, gather_mode=0, lds_addr, global_addr, type=2
  u32x4 g0 = { 0x1u,
               lds_byte,
               (uint32_t)ga,
               ((uint32_t)(ga >> 32) & 0x01FFFFFFu) | 0x80000000u /* type=2 */ };
  // Group 1: wg_mask=0, data_size=2 (4B), no atomic-barrier/iterate/pad.
  // tensor_dim0/1 set huge so no OOB zeroing (tiles are always fully in-bounds).
  const uint32_t td0 = 0x7FFFFFFFu, td1 = 0x7FFFFFFFu;
  u32x8 g1 = { (2u << 16),                                   // data_size
               (td0 & 0xFFFFu) << 16,                        // abar_addr=0 | tensor_dim0.lo
               (td0 >> 16) | ((td1 & 0xFFFFu) << 16),        // tensor_dim0.hi | tensor_dim1.lo
               (td1 >> 16) | (tile_d0 << 16),                // tensor_dim1.hi | tile_dim0
               tile_d1,                                      // tile_dim1 | tile_dim2=0
               (uint32_t)stride_elems,                       // tensor_dim0_stride[31:0]
               (uint32_t)(stride_elems >> 32) & 0xFFFFu,     // stride[47:32] | dim1_stride.lo=0
               0u };
  u32x4 zz = { 0u, 0u, 0u, 0u };                             // groups 2/3: all tile dims unused
  asm volatile("tensor_load_to_lds %0, %1, %2, %3 th:TH_LOAD_NT"
               :: "s"(g0), "s"(g1), "s"(zz), "s"(zz)
               : "memory");
}

// ---------------------------------------------------------------------------
// Main compositor. 256 threads = 8 waves; each wave owns 32 pixels and stages
// its (K=16 x 32px) fragment/alpha tiles into LDS via two TDM descriptors.
// ---------------------------------------------------------------------------
__global__ void AlphaCompositor_49349174231582_kernel(
    const int*   __restrict__ frag,    // (N,K,H,W)
    const float* __restrict__ alpha,   // (N,K,H,W)
    const float* __restrict__ ptclds,  // (C,P) channel-major (fallback path)
    const float4* __restrict__ pts4,   // (P) interleaved RGBA (fast path)
    const float* __restrict__ bgc,     // (3)
    float*       __restrict__ out,     // (N,C,H,W)
    int use_t, int pcount) {
  __shared__ int   s_frag [8 * K_ * 32];   // per-wave slice: 16x32 ints
  __shared__ float s_alpha[8 * K_ * 32];

  const int tid  = threadIdx.x;
  const int lane = tid & 31;
  const int wave = __builtin_amdgcn_readfirstlane(tid >> 5);   // SGPR wave id

  const int blocksPerImg = HW_ / 256;
  const int n    = blockIdx.x / blocksPerImg;
  const int pix0 = (blockIdx.x % blocksPerImg) * 256;
  const int wavePix = pix0 + wave * 32;                        // scalar
  const size_t gbase = (size_t)n * (K_ * HW_) + (size_t)wavePix;

  // Two async TDM tile loads per wave (in-order on TENSORcnt).
  tdm_load_tile_2d((uint32_t)(uintptr_t)&s_frag [wave * (K_ * 32)],
                   frag  + gbase, 32u, (uint32_t)K_, (uint64_t)HW_);
  tdm_load_tile_2d((uint32_t)(uintptr_t)&s_alpha[wave * (K_ * 32)],
                   alpha + gbase, 32u, (uint32_t)K_, (uint64_t)HW_);
  __builtin_amdgcn_s_wait_tensorcnt(0);     // wave reads only its own slice -> no barrier

  const int sb = wave * (K_ * 32) + lane;
  const int f0 = s_frag[sb];

  float trans = 1.0f;
  float ax = 0.f, ay = 0.f, az = 0.f, aw = 0.f;
#pragma unroll
  for (int k = 0; k < K_; ++k) {
    const int   f = s_frag [sb + k * 32];
    const float a = (f >= 0) ? s_alpha[sb + k * 32] : 0.0f;  // invalid slot: weight 0
    const int idx = (f >= 0) ? f : 0;
    const float w = a * trans;
    float cx, cy, cz, cw;
    if (use_t) {
      float4 c = pts4[idx];          // single b128 gather, L2-resident table
      cx = c.x; cy = c.y; cz = c.z; cw = c.w;
    } else {
      cx = ptclds[idx];
      cy = ptclds[pcount + idx];
      cz = ptclds[2 * pcount + idx];
      cw = ptclds[3 * pcount + idx];
    }
    ax = fmaf(w, cx, ax);
    ay = fmaf(w, cy, ay);
    az = fmaf(w, cz, az);
    aw = fmaf(w, cw, aw);
    trans *= (1.0f - a);
  }

  if (f0 < 0) {            // pixel with no points: background RGBA
    ax = bgc[0]; ay = bgc[1]; az = bgc[2]; aw = 1.0f;
  }

  // Non-temporal stores: output is write-once, never re-read on device.
  const size_t o = (size_t)n * (4 * HW_) + (size_t)(wavePix + lane);
  __builtin_nontemporal_store(ax, &out[o]);
  __builtin_nontemporal_store(ay, &out[o + HW_]);
  __builtin_nontemporal_store(az, &out[o + 2 * HW_]);
  __builtin_nontemporal_store(aw, &out[o + 3 * HW_]);
}

// ---------------------------------------------------------------------------
// Pre-pass: transpose ptclds (C,P) channel-major -> float4[P] so each gather
// is a single b128 load. NT loads (read-once source), RT stores so the
// interleaved table lands in L2 and stays resident for the gathers.
// ---------------------------------------------------------------------------
__global__ void ptclds_transpose_kernel(const float* __restrict__ pt,
                                        float4* __restrict__ out4, int pcount) {
  int i = blockIdx.x * blockDim.x + threadIdx.x;
  if (i < pcount) {
    float4 v;
    v.x = __builtin_nontemporal_load(&pt[i]);
    v.y = __builtin_nontemporal_load(&pt[pcount + i]);
    v.z = __builtin_nontemporal_load(&pt[2 * pcount + i]);
    v.w = __builtin_nontemporal_load(&pt[3 * pcount + i]);
    out4[i] = v;
  }
}

extern "C" void kernel_launch(void* const* d_in, const int* in_sizes, int n_in,
                              void* d_out, int out_size, void* d_ws, size_t ws_size,
                              hipStream_t stream) {
  const int*   frag  = (const int*)  d_in[0];
  const float* alpha = (const float*)d_in[1];
  const float* pt    = (const float*)d_in[2];
  const float* bgc   = (const float*)d_in[3];
  float*       out   = (float*)d_out;

  const int pcount = in_sizes[2] / 4;                       // (C=4, P)
  const bool use_t = ws_size >= (size_t)pcount * sizeof(float4);
  float4* pts4 = (float4*)d_ws;

  if (use_t) {
    ptclds_transpose_kernel<<<(pcount + 255) / 256, 256, 0, stream>>>(pt, pts4, pcount);
  }
  const int nblocks = (N_ * HW_) / 256;                     // 8192 blocks, 8 waves each
  AlphaCompositor_49349174231582_kernel<<<nblocks, 256, 0, stream>>>(
      frag, alpha, pt, pts4, bgc, out, use_t ? 1 : 0, pcount);
}